// DSHWModule_63556926046422
// MI455X (gfx1250) — compile-verified
//
#include <hip/hip_runtime.h>

// Double-seasonal Holt-Winters, multiplicative, enable_ar=False.
// 1024 independent series (bs=64, f=16), sequential 512-step recurrence each.
// One thread per series; seasonal state in LDS, [phase][tid] layout (bank-conflict-free
// since all lanes of a wave are at the same phase). Serial s/t chain is ~2 FMA/step;
// all divisions (rcp + 1 Newton step ~= 0.5ulp) are off the critical chain.

#define P1    24
#define P2    168
#define NSTEP 512
#define FEAT  16
#define BSZ   64
#define MAXH  336
#define TPB   64            // 4 batches x 16 features per block
#define BPB   (TPB / FEAT)  // batches per block

// d_out offsets (floats), reference return order: fcast,yhat,e,Ic,wc,t,s
#define OFF_FC 0
#define OFF_YH (OFF_FC + BSZ * MAXH * FEAT)   // 344064
#define OFF_E  (OFF_YH + BSZ * NSTEP * FEAT)  // 868352
#define OFF_IC (OFF_E  + BSZ * NSTEP * FEAT)  // 1392640
#define OFF_WC (OFF_IC + BSZ * P1 * FEAT)     // 1417216
#define OFF_T  (OFF_WC + BSZ * P2 * FEAT)     // 1589248
#define OFF_S  (OFF_T  + BSZ * FEAT)          // 1590272

// v_rcp_f32 (~1 ulp) + one Newton-Raphson step -> ~0.5 ulp, matches IEEE div
// closely enough for a 512-step multiplicative recurrence.
__device__ __forceinline__ float frcp(float x) {
    float r = __builtin_amdgcn_rcpf(x);
    return fmaf(fmaf(-x, r, 1.0f), r, r);
}
__device__ __forceinline__ float sigmoidf(float x) { return frcp(1.0f + __expf(-x)); }

__global__ __launch_bounds__(TPB) void dshw_kernel(
    const float* __restrict__ y,
    const float* __restrict__ alphas,
    const float* __restrict__ betas,
    const float* __restrict__ gammas,
    const float* __restrict__ omegas,
    float* __restrict__ out)
{
    __shared__ float IcL[P1 * TPB];   // [phase][tid]
    __shared__ float wcL[P2 * TPB];   // [phase][tid]

    const int tid = threadIdx.x;
    const int f   = tid & (FEAT - 1);
    const int b   = blockIdx.x * BPB + (tid >> 4);

    const float* Y = y + (size_t)b * NSTEP * FEAT + f;   // Y[i*FEAT] = y[b,i,f]

    const float a  = sigmoidf(alphas[f]);
    const float bb = sigmoidf(betas[f]);
    const float g  = sigmoidf(gammas[f]);
    const float o  = sigmoidf(omegas[f]);

    // ---------------- init params (one pass over y[b, 0:336, f]) ----------------
    // I1 raw: y[p] + y[p+24], p<24 ; normalized by 24/sum48  (== (raw/2)/mean48)
    float sum48 = 0.0f;
    for (int p = 0; p < P1; ++p) {
        float va = Y[p * FEAT];
        float vb = Y[(p + P1) * FEAT];
        IcL[p * TPB + tid] = va + vb;
        sum48 += va + vb;
    }
    // w raw: y[p] + y[p+168], p<168 ; plus range sums for trend/level init
    float sumA = 0.0f, sumB = 0.0f, y0 = 0.0f, y167 = 0.0f;
    for (int p = 0; p < P2; ++p) {
        float va = Y[p * FEAT];
        float vb = Y[(p + P2) * FEAT];
        wcL[p * TPB + tid] = va + vb;
        sumA += va;
        sumB += vb;
        if (p == 0)      y0   = va;
        if (p == P2 - 1) y167 = va;
    }
    const float sum336 = sumA + sumB;

    const float scale1 = 24.0f * frcp(sum48);    // I1 = raw * 24/sum48
    for (int p = 0; p < P1; ++p)
        IcL[p * TPB + tid] *= scale1;

    const float scale2 = 168.0f * frcp(sum336);  // sidx2 = raw * 168/sum336
    {
        int pm = 0;
        for (int p = 0; p < P2; ++p) {
            wcL[p * TPB + tid] = wcL[p * TPB + tid] * scale2 * frcp(IcL[pm * TPB + tid]);
            pm = (pm == P1 - 1) ? 0 : pm + 1;
        }
    }
    // t0 = ( (meanA - meanB)/P2 + (y167 - y0)/P2 ) / 2 ; s0 = mean336 - (P2+0.5)*t0
    float t = ((sumA - sumB) * (1.0f / (168.0f * 168.0f)) +
               (y167 - y0)   * (1.0f / 168.0f)) * 0.5f;
    float s = sum336 * (1.0f / 336.0f) - 168.5f * t;

    // ---------------- sequential scan, 512 steps ----------------
    float* out_yh = out + OFF_YH + (size_t)b * NSTEP * FEAT + f;
    float* out_e  = out + OFF_E  + (size_t)b * NSTEP * FEAT + f;

    int i1 = 0, i2 = 0;
    for (int c = 0; c < NSTEP; c += 8) {
        float yv[8];
        #pragma unroll
        for (int k = 0; k < 8; ++k)
            yv[k] = Y[(c + k) * FEAT];
        if (c + 8 < NSTEP)
            __builtin_prefetch(&Y[(c + 8) * FEAT], 0, 0);   // global_prefetch_b8

        #pragma unroll
        for (int k = 0; k < 8; ++k) {
            const float yt  = yv[k];
            const float Icv = IcL[i1 * TPB + tid];   // written >=24 steps ago
            const float wcv = wcL[i2 * TPB + tid];   // written >=168 steps ago

            const float st   = s + t;
            const float prod = Icv * wcv;
            const float yh   = st * prod;
            const float q    = yt * frcp(prod);          // off critical chain
            const float snew = fmaf(a, q - st, st);      // a*q + (1-a)*st
            const float tnew = fmaf(bb, snew - st, t);   // b*(snew-s) + (1-b)*t
            const float u    = q * frcp(snew);           // yt/(Icv*wcv*snew)

            // Ic' = g*(yt/(snew*wcv)) + (1-g)*Icv = Icv + g*(u*Icv - Icv)
            IcL[i1 * TPB + tid] = fmaf(g, fmaf(u, Icv, -Icv), Icv);
            // wc' = o*(yt/(snew*Icv)) + (1-o)*wcv = wcv + o*(u*wcv - wcv)
            wcL[i2 * TPB + tid] = fmaf(o, fmaf(u, wcv, -wcv), wcv);

            out_yh[(c + k) * FEAT] = yh;
            out_e [(c + k) * FEAT] = yt - yh;

            s = snew; t = tnew;
            i1 = (i1 == P1 - 1) ? 0 : i1 + 1;
            i2 = (i2 == P2 - 1) ? 0 : i2 + 1;
        }
    }

    // ---------------- forecast (roll folded into index: (512+j) % P) ----------------
    float* out_fc = out + OFF_FC + (size_t)b * MAXH * FEAT + f;
    {
        int j1 = NSTEP % P1;   // 8
        int j2 = NSTEP % P2;   // 8
        for (int j = 0; j < MAXH; ++j) {
            const float h  = (float)(j + 1);
            const float fc = fmaf(h, t, s) * IcL[j1 * TPB + tid] * wcL[j2 * TPB + tid];
            out_fc[j * FEAT] = fc;
            j1 = (j1 == P1 - 1) ? 0 : j1 + 1;
            j2 = (j2 == P2 - 1) ? 0 : j2 + 1;
        }
    }

    // ---------------- rolled state outputs + scalars ----------------
    float* out_Ic = out + OFF_IC + (size_t)b * P1 * FEAT + f;
    float* out_wc = out + OFF_WC + (size_t)b * P2 * FEAT + f;
    {
        int j1 = NSTEP % P1;   // rolled[p] = Ic[(p + 512) % 24]
        for (int p = 0; p < P1; ++p) {
            out_Ic[p * FEAT] = IcL[j1 * TPB + tid];
            j1 = (j1 == P1 - 1) ? 0 : j1 + 1;
        }
        int j2 = NSTEP % P2;   // rolled[p] = wc[(p + 512) % 168]
        for (int p = 0; p < P2; ++p) {
            out_wc[p * FEAT] = wcL[j2 * TPB + tid];
            j2 = (j2 == P2 - 1) ? 0 : j2 + 1;
        }
    }
    out[OFF_T + b * FEAT + f] = t;
    out[OFF_S + b * FEAT + f] = s;
}

extern "C" void kernel_launch(void* const* d_in, const int* in_sizes, int n_in,
                              void* d_out, int out_size, void* d_ws, size_t ws_size,
                              hipStream_t stream) {
    (void)in_sizes; (void)n_in; (void)out_size; (void)d_ws; (void)ws_size;
    const float* y      = (const float*)d_in[0];
    const float* alphas = (const float*)d_in[1];
    const float* betas  = (const float*)d_in[2];
    const float* gammas = (const float*)d_in[3];
    const float* omegas = (const float*)d_in[4];
    // d_in[5] = phis : unused (enable_ar=False path in reference)
    float* out = (float*)d_out;

    dshw_kernel<<<BSZ / BPB, TPB, 0, stream>>>(y, alphas, betas, gammas, omegas, out);
}